// PoseGraph_4320737100582
// MI455X (gfx1250) — compile-verified
//
#include <hip/hip_runtime.h>
#include <cstdint>

#define TPB 256

// ---- CDNA5 async global<->LDS path (gfx1250), with safe fallback ----------
#if defined(__has_builtin)
#  if __has_builtin(__builtin_amdgcn_global_load_async_to_lds_b32)
#    define HAVE_ASYNC_LOAD 1
#  endif
#  if __has_builtin(__builtin_amdgcn_global_store_async_from_lds_b32)
#    define HAVE_ASYNC_STORE 1
#  endif
#  if __has_builtin(__builtin_amdgcn_s_wait_asynccnt)
#    define HAVE_WAIT_ASYNC 1
#  endif
#endif

// Builtin params are int* with target address spaces: as(1) for global, as(3) for LDS.
typedef __attribute__((address_space(3))) int lds_int_t;
typedef __attribute__((address_space(1))) int glob_int_t;

// generic LDS pointer = {aperture_hi32, lds_offset32}; low 32 bits are the as(3) offset
__device__ __forceinline__ lds_int_t* to_lds(const void* p) {
  return (lds_int_t*)(uint32_t)(uintptr_t)p;
}
__device__ __forceinline__ glob_int_t* to_glob(const void* p) {
  return (glob_int_t*)(uintptr_t)p;
}

__device__ __forceinline__ void async_copy_g2l(const float* g, float* l) {
#if defined(HAVE_ASYNC_LOAD)
  __builtin_amdgcn_global_load_async_to_lds_b32(to_glob(g), to_lds(l), 0, 0);
#else
  *l = *g;
#endif
}
__device__ __forceinline__ void async_copy_l2g(float* g, const float* l) {
#if defined(HAVE_ASYNC_STORE)
  __builtin_amdgcn_global_store_async_from_lds_b32(to_glob(g), to_lds(l), 0, 0);
#else
  *g = *l;
#endif
}
__device__ __forceinline__ void wait_async0() {
#if defined(HAVE_WAIT_ASYNC)
  __builtin_amdgcn_s_wait_asynccnt(0);
#else
  asm volatile("s_wait_asynccnt 0" ::: "memory");
#endif
}

// ---- small SE3 math helpers (f32 scalar VALU) -----------------------------
struct V3 { float x, y, z; };
__device__ __forceinline__ V3 operator+(V3 a, V3 b) { return {a.x + b.x, a.y + b.y, a.z + b.z}; }
__device__ __forceinline__ V3 operator-(V3 a, V3 b) { return {a.x - b.x, a.y - b.y, a.z - b.z}; }
__device__ __forceinline__ V3 operator*(float s, V3 a) { return {s * a.x, s * a.y, s * a.z}; }
__device__ __forceinline__ float dot3(V3 a, V3 b) {
  return fmaf(a.x, b.x, fmaf(a.y, b.y, a.z * b.z));
}
__device__ __forceinline__ V3 cross3(V3 a, V3 b) {
  return { fmaf(a.y, b.z, -a.z * b.y),
           fmaf(a.z, b.x, -a.x * b.z),
           fmaf(a.x, b.y, -a.y * b.x) };
}
// t + 2*cross(v, cross(v,t) + w*t)
__device__ __forceinline__ V3 qrot(V3 v, float w, V3 t) {
  V3 u = cross3(v, t) + w * t;
  return t + 2.0f * cross3(v, u);
}
struct Q { V3 v; float w; };
__device__ __forceinline__ Q qmul(Q a, Q b) {
  Q r;
  r.w = a.w * b.w - dot3(a.v, b.v);
  r.v = (a.w * b.v) + (b.w * a.v) + cross3(a.v, b.v);
  return r;
}

// ---------------------------------------------------------------------------
__global__ void __launch_bounds__(TPB)
posegraph_se3log_kernel(const float* __restrict__ nodes,
                        const int*   __restrict__ edges,
                        const float* __restrict__ poses,
                        float*       __restrict__ out,
                        int E) {
  __shared__ float lds[TPB * 7];

  const int t     = threadIdx.x;
  const int base  = blockIdx.x * TPB;
  const int rem   = E - base;
  const int valid = rem < TPB ? rem : TPB;
  const bool full = (valid == TPB);   // true for every block when E % TPB == 0

  // ---- Stage poses tile (valid*7 contiguous floats) global -> LDS --------
  // Full-tile fast path: unguarded -> single async clause, immediate offsets.
  const float* src = poses + (long long)base * 7;
  if (full) {
#pragma unroll
    for (int k = 0; k < 7; ++k) {
      const int idx = t + k * TPB;
      async_copy_g2l(src + idx, &lds[idx]);
    }
  } else {
    const int nfl = valid * 7;
#pragma unroll
    for (int k = 0; k < 7; ++k) {
      const int idx = t + k * TPB;
      if (idx < nfl) async_copy_g2l(src + idx, &lds[idx]);
    }
  }

  // ---- Overlap: kick off random node gathers while the pose DMA flies ----
  const bool active = t < valid;
  float a1[7], a2[7];
  if (active) {
    const int2 pr = ((const int2*)edges)[base + t];    // one b64 load
    const float* n1 = nodes + (long long)pr.x * 7;     // L2-resident (5.6 MB set)
    const float* n2 = nodes + (long long)pr.y * 7;
#pragma unroll
    for (int k = 0; k < 7; ++k) a1[k] = n1[k];
#pragma unroll
    for (int k = 0; k < 7; ++k) a2[k] = n2[k];
  }

  wait_async0();      // pose tile landed in LDS
  __syncthreads();

  float pose[7];
  if (active) {
#pragma unroll
    for (int k = 0; k < 7; ++k) pose[k] = lds[t * 7 + k];  // stride 7: conflict-free
  }
  __syncthreads();    // poses consumed; LDS reused for output staging

  if (active) {
    V3 t1 = {a1[0], a1[1], a1[2]};  Q q1 = {{a1[3], a1[4], a1[5]}, a1[6]};
    V3 t2 = {a2[0], a2[1], a2[2]};  Q q2 = {{a2[3], a2[4], a2[5]}, a2[6]};
    V3 tp = {pose[0], pose[1], pose[2]};
    Q  qp = {{pose[3], pose[4], pose[5]}, pose[6]};

    // A = inv(node1)
    Q  qA = {-1.0f * q1.v, q1.w};
    V3 tA = -1.0f * qrot(qA.v, qA.w, t1);
    // B = A * node2
    V3 tB = tA + qrot(qA.v, qA.w, t2);
    Q  qB = qmul(qA, q2);
    // C = inv(pose)
    Q  qC = {-1.0f * qp.v, qp.w};
    V3 tC = -1.0f * qrot(qC.v, qC.w, tp);
    // err = B * C
    V3 tE = tB + qrot(qB.v, qB.w, tC);
    Q  qE = qmul(qB, qC);

    // se3_log(err)
    float sgn = (qE.w < 0.0f) ? -1.0f : 1.0f;
    V3 v = sgn * qE.v;
    float w = sgn * qE.w;
    float n = sqrtf(dot3(v, v));
    float theta = 2.0f * atan2f(n, w);
    const float eps = 1e-8f;
    float factor = (n < eps) ? (2.0f / fmaxf(w, eps)) : (theta / fmaxf(n, eps));
    V3 phi = factor * v;
    float th2 = theta * theta;
    bool tiny = theta < 1e-4f;
    float safe = tiny ? 1.0f : theta;
    float c = tiny ? (1.0f / 12.0f + th2 * (1.0f / 720.0f))
                   : (1.0f / fmaxf(th2, eps) -
                      (1.0f + cosf(theta)) / (2.0f * safe * sinf(safe)));
    V3 pxt = cross3(phi, tE);
    V3 tau = tE - 0.5f * pxt + c * cross3(phi, pxt);

    lds[t * 6 + 0] = tau.x;
    lds[t * 6 + 1] = tau.y;
    lds[t * 6 + 2] = tau.z;
    lds[t * 6 + 3] = phi.x;
    lds[t * 6 + 4] = phi.y;
    lds[t * 6 + 5] = phi.z;
  }
  __syncthreads();

  // ---- Coalesced LDS -> global output (async DMA, ASYNCcnt-tracked) ------
  float* dst = out + (long long)base * 6;
  if (full) {
#pragma unroll
    for (int k = 0; k < 6; ++k) {
      const int idx = t + k * TPB;
      async_copy_l2g(dst + idx, &lds[idx]);
    }
  } else {
    const int nof = valid * 6;
#pragma unroll
    for (int k = 0; k < 6; ++k) {
      const int idx = t + k * TPB;
      if (idx < nof) async_copy_l2g(dst + idx, &lds[idx]);
    }
  }
#if defined(HAVE_ASYNC_STORE)
  wait_async0();  // LDS must stay valid until the async stores complete
#endif
}

// ---------------------------------------------------------------------------
extern "C" void kernel_launch(void* const* d_in, const int* in_sizes, int n_in,
                              void* d_out, int out_size, void* d_ws, size_t ws_size,
                              hipStream_t stream) {
  const float* nodes = (const float*)d_in[0];   // N*7 f32
  const int*   edges = (const int*)d_in[1];     // E*2 i32
  const float* poses = (const float*)d_in[2];   // E*7 f32
  float* out = (float*)d_out;                   // E*6 f32

  const int E = in_sizes[1] / 2;
  const int blocks = (E + TPB - 1) / TPB;
  hipLaunchKernelGGL(posegraph_se3log_kernel, dim3(blocks), dim3(TPB), 0, stream,
                     nodes, edges, poses, out, E);
}